// LindBladEvolve_46608985096954
// MI455X (gfx1250) — compile-verified
//
#include <hip/hip_runtime.h>
#include <hip/hip_bf16.h>
#include <math.h>

// ---------------------------------------------------------------------------
// Problem constants (match reference)
// ---------------------------------------------------------------------------
#define BQ    8192
#define TQ    2048
#define NSTEP (TQ - 1)      // 2047
#define NCQ   4
#define NJQ   3
#define DTQ   0.01f
#define EPSF  1e-30f

typedef float v2f __attribute__((ext_vector_type(2)));
typedef float v4f __attribute__((ext_vector_type(4)));
typedef float v8f __attribute__((ext_vector_type(8)));

// ---------------------------------------------------------------------------
// CDNA5 helpers
// ---------------------------------------------------------------------------
// D = A(16x4) * B(4x16) + C, all fp32, wave32 WMMA
static __device__ __forceinline__ v8f wmma4(v2f a, v2f b, v8f c) {
  return __builtin_amdgcn_wmma_f32_16x16x4_f32(
      /*neg_a=*/false, a, /*neg_b=*/false, b,
      /*c_mod=*/(short)0, c, /*reuse_a=*/false, /*reuse_b=*/false);
}

// lane <-> lane^16 exchange (ds_swizzle group-of-32: xor=0x10, and=0x1f)
static __device__ __forceinline__ float swapx16(float v) {
  return __int_as_float(__builtin_amdgcn_ds_swizzle(__float_as_int(v), 0x401F));
}
// sum over both half-waves of a per-lane partial -> valid in ALL lanes
static __device__ __forceinline__ float allsum(float p) { return p + swapx16(p); }

// Convert a 16x16 state from WMMA C/D layout (xd[r]: lane<16 -> row r,
// lane>=16 -> row r+8, col = lane&15) into B-operand layout for the next
// multiply: chunk c VGPR0 = rows 4c (lo) / 4c+2 (hi), VGPR1 = 4c+1 / 4c+3.
static __device__ __forceinline__ void d2b(const float xd[8], v2f xb[4], bool lo) {
  float s[8];
#pragma unroll
  for (int r = 0; r < 8; ++r) s[r] = swapx16(xd[r]);
  xb[0] = (v2f){lo ? xd[0] : s[2], lo ? xd[1] : s[3]};
  xb[1] = (v2f){lo ? xd[4] : s[6], lo ? xd[5] : s[7]};
  xb[2] = (v2f){lo ? s[0] : xd[2], lo ? s[1] : xd[3]};
  xb[3] = (v2f){lo ? s[4] : xd[6], lo ? s[5] : xd[7]};
}

// Real 16x16 embedding of complex 8x8:  E = [[Re, -Im], [Im, Re]]
static __device__ __forceinline__ float cplx_embed(float re, float im, int row, int col) {
  if (row < 8) return (col < 8) ? re : -im;
  return (col < 8) ? im : re;
}

// ---------------------------------------------------------------------------
// Kernel 1: embed jump operators c_j into WMMA-A chunk layout
//   layout: ECOP[j*256 + lane*8 + c*2 + vgpr],  A(row=lane&15,
//           col = 4c + vgpr + 2*(lane>=16))
// ---------------------------------------------------------------------------
__global__ __launch_bounds__(64) void build_cops_kernel(
    const float* __restrict__ copre, const float* __restrict__ copim,
    float* __restrict__ ECOP) {
  const int j = blockIdx.x;
  for (int e = threadIdx.x; e < 256; e += 64) {
    int lane = e >> 3, q = e & 7, c = q >> 1, jj = q & 1;
    int row = lane & 15;
    int col = 4 * c + jj + ((lane >= 16) ? 2 : 0);
    int rr = row & 7, cc = col & 7;
    float re = copre[j * 64 + rr * 8 + cc];
    float im = copim[j * 64 + rr * 8 + cc];
    ECOP[j * 256 + e] = cplx_embed(re, im, row, col);
  }
}

// ---------------------------------------------------------------------------
// Kernel 2: per-step Hamiltonian H_k, effective propagator
//   P_k = expm(-i*DT*(H_k - 0.5i*S)) via 12-term Taylor (||M|| ~ 0.1),
// written as real 16x16 embeddings in WMMA-A chunk layout.
// One 64-thread block per step k; thread t owns entry (i,l)=(t/8,t%8).
// ---------------------------------------------------------------------------
__global__ __launch_bounds__(64) void build_prop_kernel(
    const float* __restrict__ h0re, const float* __restrict__ h0im,
    const float* __restrict__ chre, const float* __restrict__ chim,
    const float* __restrict__ cr,   const float* __restrict__ ci,
    const float* __restrict__ copre, const float* __restrict__ copim,
    float* __restrict__ EP, float* __restrict__ EH) {
  const int k = blockIdx.x;            // 0 .. NSTEP-1
  const int t = threadIdx.x;           // 0 .. 63
  const int i = t >> 3, l = t & 7;

  // Hermitian part of H0
  float hre = 0.5f * (h0re[i * 8 + l] + h0re[l * 8 + i]);
  float him = 0.5f * (h0im[i * 8 + l] - h0im[l * 8 + i]);
#pragma unroll
  for (int j = 0; j < NCQ; ++j) {
    float a = cr[k * NCQ + j];
    hre += a * 0.5f * (chre[j * 64 + i * 8 + l] + chre[j * 64 + l * 8 + i]);
    him += a * 0.5f * (chim[j * 64 + i * 8 + l] - chim[j * 64 + l * 8 + i]);
    // + i * ci * Hi  (Hi real antisymmetric) -> imaginary part
    float b = ci[k * NCQ + j];
    him += b * 0.5f * (chre[j * 64 + i * 8 + l] - chre[j * 64 + l * 8 + i]);
  }

  // S[i][l] = sum_j sum_b conj(c_j[b][i]) * c_j[b][l]
  float sre = 0.f, sim = 0.f;
#pragma unroll
  for (int j = 0; j < NJQ; ++j)
    for (int b = 0; b < 8; ++b) {
      float ar = copre[j * 64 + b * 8 + i], ai = copim[j * 64 + b * 8 + i];
      float br = copre[j * 64 + b * 8 + l], bi = copim[j * 64 + b * 8 + l];
      sre += ar * br + ai * bi;
      sim += ar * bi - ai * br;
    }

  // M = -i*DT*H - 0.5*DT*S
  float mre =  DTQ * him - 0.5f * DTQ * sre;
  float mim = -DTQ * hre - 0.5f * DTQ * sim;

  __shared__ float Mre[64], Mim[64], Are[64], Aim[64], Bre[64], Bim[64];
  Mre[t] = mre; Mim[t] = mim;
  float tre = (i == l) ? 1.f : 0.f;          // term = I
  float pre = tre, pim = 0.f;                // P accumulates I + ...
  Are[t] = tre; Aim[t] = 0.f;
  __syncthreads();
#pragma unroll 1
  for (int n = 1; n <= 12; ++n) {            // Taylor: term *= M / n
    const float* Tr = (n & 1) ? Are : Bre;
    const float* Ti = (n & 1) ? Aim : Bim;
    float nr = 0.f, ni = 0.f;
#pragma unroll
    for (int q = 0; q < 8; ++q) {
      float ar = Tr[i * 8 + q], ai = Ti[i * 8 + q];
      float br = Mre[q * 8 + l], bi = Mim[q * 8 + l];
      nr += ar * br - ai * bi;
      ni += ar * bi + ai * br;
    }
    float s = 1.0f / (float)n;
    nr *= s; ni *= s;
    pre += nr; pim += ni;
    __syncthreads();
    float* Wr = (n & 1) ? Bre : Are;
    float* Wi = (n & 1) ? Bim : Aim;
    Wr[t] = nr; Wi[t] = ni;
    __syncthreads();
  }

  // stash P and H for arbitrary-(row,col) layout writes
  Mre[t] = pre; Mim[t] = pim;
  Are[t] = hre; Aim[t] = him;
  __syncthreads();
  for (int e = t; e < 256; e += 64) {
    int lane = e >> 3, q = e & 7, c = q >> 1, jj = q & 1;
    int row = lane & 15;
    int col = 4 * c + jj + ((lane >= 16) ? 2 : 0);
    int rr = row & 7, cc = col & 7;
    EP[k * 256 + e] = cplx_embed(Mre[rr * 8 + cc], Mim[rr * 8 + cc], row, col);
    EH[k * 256 + e] = cplx_embed(Are[rr * 8 + cc], Aim[rr * 8 + cc], row, col);
  }
}

// ---------------------------------------------------------------------------
// Kernel 3: trajectory scan.  One wave32 = 16 trajectories (N dim of WMMA).
// Per step: 4 WMMA propagate + 12 WMMA jump candidates + 4 WMMA energy.
// ---------------------------------------------------------------------------
__global__ __launch_bounds__(128) void traj_kernel(
    const float* __restrict__ EP, const float* __restrict__ EH,
    const float* __restrict__ ECOP,
    const float* __restrict__ init_diag,
    const float* __restrict__ u_reset, const float* __restrict__ u_jump,
    float* __restrict__ out) {
  const int lane = threadIdx.x & 31;
  const int wave = blockIdx.x * (blockDim.x >> 5) + (threadIdx.x >> 5);
  const int traj = wave * 16 + (lane & 15);
  const bool lo = lane < 16;

  // jump-operator A operands (loop invariant, kept in VGPRs)
  v2f copA[NJQ][4];
#pragma unroll
  for (int j = 0; j < NJQ; ++j) {
    const v4f* p = reinterpret_cast<const v4f*>(ECOP + j * 256 + lane * 8);
    v4f q0 = p[0], q1 = p[1];
    copA[j][0] = (v2f){q0.x, q0.y}; copA[j][1] = (v2f){q0.z, q0.w};
    copA[j][2] = (v2f){q1.x, q1.y}; copA[j][3] = (v2f){q1.z, q1.w};
  }

  // psi0 = sqrt(init_diag)/||.||  -> D-layout (rows 0..7 = Re, 8..15 = Im = 0)
  float nrm2 = 0.f, dg[8];
#pragma unroll
  for (int m = 0; m < 8; ++m) { dg[m] = init_diag[m]; nrm2 += dg[m]; }
  float isc = rsqrtf(nrm2);
  float xn[8];
#pragma unroll
  for (int r = 0; r < 8; ++r) xn[r] = lo ? sqrtf(dg[r]) * isc : 0.f;
  v2f xb[4];
  d2b(xn, xb, lo);

  float rth = u_reset[traj];
  float logp = 0.f, energy = 0.f;

#pragma unroll 1
  for (int k = 0; k < NSTEP; ++k) {
    const v4f* pe = reinterpret_cast<const v4f*>(EP + (size_t)k * 256 + lane * 8);
    v4f e0 = pe[0], e1 = pe[1];
    const v4f* ph = reinterpret_cast<const v4f*>(EH + (size_t)k * 256 + lane * 8);
    v4f h0 = ph[0], h1 = ph[1];
    float ur = u_reset[(size_t)(k + 1) * BQ + traj];
    float uj = u_jump[(size_t)k * BQ + traj];

    // ---- psi = P_k psi  (16x16x16 fp32 GEMM = 4 chained WMMA) ----
    v8f acc = {0.f, 0.f, 0.f, 0.f, 0.f, 0.f, 0.f, 0.f};
    acc = wmma4((v2f){e0.x, e0.y}, xb[0], acc);
    acc = wmma4((v2f){e0.z, e0.w}, xb[1], acc);
    acc = wmma4((v2f){e1.x, e1.y}, xb[2], acc);
    acc = wmma4((v2f){e1.z, e1.w}, xb[3], acc);
    float xp[8], pn = 0.f;
#pragma unroll
    for (int r = 0; r < 8; ++r) { xp[r] = acc[r]; pn += acc[r] * acc[r]; }
    float n2 = allsum(pn);                       // ||psi||^2, all lanes

    v2f xpb[4];
    d2b(xp, xpb, lo);                            // B-layout for jump matvecs

    // ---- candidates y_j = c_j psi ;  jp_j = ||y_j||^2 ----
    float y[NJQ][8], nj[NJQ];
#pragma unroll
    for (int j = 0; j < NJQ; ++j) {
      v8f a = {0.f, 0.f, 0.f, 0.f, 0.f, 0.f, 0.f, 0.f};
      a = wmma4(copA[j][0], xpb[0], a);
      a = wmma4(copA[j][1], xpb[1], a);
      a = wmma4(copA[j][2], xpb[2], a);
      a = wmma4(copA[j][3], xpb[3], a);
      float pj = 0.f;
#pragma unroll
      for (int r = 0; r < 8; ++r) { y[j][r] = a[r]; pj += a[r] * a[r]; }
      nj[j] = allsum(pj);
    }

    // ---- inverse-CDF multinomial + collapse ----
    float tot = nj[0] + nj[1] + nj[2];
    float inv = 1.f / fmaxf(tot, EPSF);
    bool  s0 = uj < nj[0] * inv;
    bool  s1 = uj < (nj[0] + nj[1]) * inv;
    float nsel = s0 ? nj[0] : (s1 ? nj[1] : nj[2]);
    float jscale = 1.f / fmaxf(sqrtf(nsel), EPSF);
    bool  jump = (n2 <= rth) || (n2 < 1e-6f);

#pragma unroll
    for (int r = 0; r < 8; ++r) {
      float ys = s0 ? y[0][r] : (s1 ? y[1][r] : y[2][r]);
      xn[r] = jump ? ys * jscale : xp[r];
    }
    logp += jump ? __logf(fmaxf(1.f - n2, EPSF)) : __logf(fmaxf(n2, EPSF));
    rth = jump ? ur : rth;

    d2b(xn, xb, lo);                             // state for next step

    // ---- energy += psi^T E(H_k) psi ----
    v8f w = {0.f, 0.f, 0.f, 0.f, 0.f, 0.f, 0.f, 0.f};
    w = wmma4((v2f){h0.x, h0.y}, xb[0], w);
    w = wmma4((v2f){h0.z, h0.w}, xb[1], w);
    w = wmma4((v2f){h1.x, h1.y}, xb[2], w);
    w = wmma4((v2f){h1.z, h1.w}, xb[3], w);
    float pd = 0.f;
#pragma unroll
    for (int r = 0; r < 8; ++r) pd += xn[r] * w[r];
    energy += allsum(pd);
  }

  if (lo) {
    out[(size_t)traj * 2 + 0] = energy;
    out[(size_t)traj * 2 + 1] = logp;
  }
}

// ---------------------------------------------------------------------------
// Host launch
// ---------------------------------------------------------------------------
extern "C" void kernel_launch(void* const* d_in, const int* in_sizes, int n_in,
                              void* d_out, int out_size, void* d_ws, size_t ws_size,
                              hipStream_t stream) {
  (void)in_sizes; (void)n_in; (void)out_size; (void)ws_size;
  const float* h0re  = (const float*)d_in[0];
  const float* h0im  = (const float*)d_in[1];
  const float* chre  = (const float*)d_in[2];
  const float* chim  = (const float*)d_in[3];
  const float* cr    = (const float*)d_in[4];
  const float* ci    = (const float*)d_in[5];
  const float* copre = (const float*)d_in[6];
  const float* copim = (const float*)d_in[7];
  const float* idiag = (const float*)d_in[8];
  const float* urst  = (const float*)d_in[9];
  const float* ujmp  = (const float*)d_in[10];

  float* EP   = (float*)d_ws;                 // NSTEP*256 floats (~2 MB)
  float* EH   = EP + (size_t)NSTEP * 256;     // NSTEP*256 floats
  float* ECOP = EH + (size_t)NSTEP * 256;     // NJ*256 floats

  build_cops_kernel<<<NJQ, 64, 0, stream>>>(copre, copim, ECOP);
  build_prop_kernel<<<NSTEP, 64, 0, stream>>>(h0re, h0im, chre, chim, cr, ci,
                                              copre, copim, EP, EH);
  // 8192 traj / 16 per wave = 512 waves; 4 waves per block -> 128 blocks
  traj_kernel<<<BQ / 64, 128, 0, stream>>>(EP, EH, ECOP, idiag, urst, ujmp,
                                           (float*)d_out);
}